// HashGridEncoder_26027501814303
// MI455X (gfx1250) — compile-verified
//
#include <hip/hip_runtime.h>
#include <cmath>
#include <cstdint>

#define NUM_LEVELS 16
#define HASHMAP_SIZE (1u << 19)
#define HASH_MASK (HASHMAP_SIZE - 1u)
#define PRIME_Y 73856093u
#define PRIME_Z 19349663u
#define SAMPLES_PER_BLOCK 16
#define BLOCK_THREADS 256

typedef float v2f __attribute__((ext_vector_type(2)));

struct ResPack { int res[NUM_LEVELS]; };

__global__ __launch_bounds__(BLOCK_THREADS)
void hashgrid_encode_kernel(const float* __restrict__ pos,     // [NS, 3]
                            const float* __restrict__ tables,  // [16, 2^19, 2]
                            float* __restrict__ out,           // [NS, 32]
                            ResPack rp, int ns)
{
    __shared__ __align__(16) float spos[SAMPLES_PER_BLOCK * 3];  // 48 floats = 192 B

    const int tid = threadIdx.x;

    // ---- Stage this block's 16 positions (192 B) into LDS with the CDNA5
    // ---- async global->LDS path (ASYNCcnt). 12 lanes x b128.
    if (tid < 12) {
        const size_t goff = (size_t)blockIdx.x * (SAMPLES_PER_BLOCK * 3) + (size_t)tid * 4;
        if (goff + 4 <= (size_t)ns * 3) {
            uint32_t lds_addr = (uint32_t)(uintptr_t)(&spos[0]) + (uint32_t)tid * 16u;
            uint64_t gaddr = (uint64_t)(uintptr_t)(pos + goff);
            asm volatile("global_load_async_to_lds_b128 %0, %1, off"
                         :: "v"(lds_addr), "v"(gaddr)
                         : "memory");
        }
    }
    asm volatile("s_wait_asynccnt 0" ::: "memory");
    __syncthreads();

    // thread = (sample-in-block, level); level is the fast dimension so a
    // wave's stores cover 256 contiguous bytes of the output.
    const int level = tid & (NUM_LEVELS - 1);
    const int sib   = tid >> 4;
    const int sample = blockIdx.x * SAMPLES_PER_BLOCK + sib;
    if (sample >= ns) return;

    const float x01 = (spos[sib * 3 + 0] + 1.0f) * 0.5f;
    const float y01 = (spos[sib * 3 + 1] + 1.0f) * 0.5f;
    const float z01 = (spos[sib * 3 + 2] + 1.0f) * 0.5f;

    const float sc = (float)(rp.res[level] - 1);
    const float sx = x01 * sc, sy = y01 * sc, sz = z01 * sc;
    const float bx = floorf(sx), by = floorf(sy), bz = floorf(sz);
    const float fx = sx - bx,  fy = sy - by,  fz = sz - bz;
    const int ix = (int)bx, iy = (int)by, iz = (int)bz;

    // int32-wraparound hash base; corner deltas just add 1 / PRIME_Y / PRIME_Z.
    const uint32_t h0 = (uint32_t)ix + (uint32_t)iy * PRIME_Y + (uint32_t)iz * PRIME_Z;

    const v2f* __restrict__ tl = (const v2f*)tables + (size_t)level * HASHMAP_SIZE;

    const float wx[2] = {1.0f - fx, fx};
    const float wy[2] = {1.0f - fy, fy};
    const float wz[2] = {1.0f - fz, fz};

    // 8 independent gathers -> compiler batches loads, single waitcnt.
    float a0 = 0.0f, a1 = 0.0f;
#pragma unroll
    for (int dx = 0; dx < 2; ++dx) {
#pragma unroll
        for (int dy = 0; dy < 2; ++dy) {
#pragma unroll
            for (int dz = 0; dz < 2; ++dz) {
                const uint32_t h = (h0 + (uint32_t)dx
                                       + (uint32_t)dy * PRIME_Y
                                       + (uint32_t)dz * PRIME_Z) & HASH_MASK;
                const v2f f = tl[h];
                const float w = wx[dx] * wy[dy] * wz[dz];
                a0 = fmaf(w, f.x, a0);
                a1 = fmaf(w, f.y, a1);
            }
        }
    }

    // Non-temporal store: 128 MB of streaming output must not evict the
    // 64 MB of hash tables that live in the 192 MB L2.
    v2f r; r.x = a0; r.y = a1;
    __builtin_nontemporal_store(r, (v2f*)(out + (size_t)sample * (NUM_LEVELS * 2) + level * 2));
}

extern "C" void kernel_launch(void* const* d_in, const int* in_sizes, int n_in,
                              void* d_out, int out_size, void* d_ws, size_t ws_size,
                              hipStream_t stream) {
    const float* pos    = (const float*)d_in[0];
    const float* tables = (const float*)d_in[1];
    float* out = (float*)d_out;

    const int ns = in_sizes[0] / 3;

    // Match numpy's double-precision resolution schedule exactly.
    ResPack rp;
    const double B = std::exp((std::log(512.0) - std::log(16.0)) / 15.0);
    for (int l = 0; l < NUM_LEVELS; ++l)
        rp.res[l] = (int)std::floor(16.0 * std::pow(B, (double)l));

    const int blocks = (ns + SAMPLES_PER_BLOCK - 1) / SAMPLES_PER_BLOCK;
    hashgrid_encode_kernel<<<blocks, BLOCK_THREADS, 0, stream>>>(pos, tables, out, rp, ns);
}